// GConvGRU_66503273611823
// MI455X (gfx1250) — compile-verified
//
#include <hip/hip_runtime.h>

#define NN  50000
#define EE  800000
#define CIN 64

typedef __attribute__((ext_vector_type(2))) float v2f;
typedef __attribute__((ext_vector_type(8))) float v8f;

// ---------------- init: Tx1 = 0, Tx2 = -X, deg = 0 ----------------
__global__ __launch_bounds__(256) void init_k(const float* __restrict__ X,
                                              float* __restrict__ Tx1,
                                              float* __restrict__ Tx2,
                                              float* __restrict__ deg) {
  int t = blockIdx.x * 256 + threadIdx.x;
  if (t < NN * CIN) {
    Tx1[t] = 0.0f;
    Tx2[t] = -X[t];
  }
  if (t < NN) deg[t] = 0.0f;
}

// ---------------- deg[src] += w ----------------
__global__ __launch_bounds__(256) void degree_k(const long long* __restrict__ src,
                                                const float* __restrict__ w,
                                                float* __restrict__ deg) {
  int e = blockIdx.x * 256 + threadIdx.x;
  if (e < EE) atomicAdd(&deg[(int)src[e]], w[e]);
}

// ---------------- deg -> d^{-1/2} in place ----------------
__global__ __launch_bounds__(256) void dinv_k(float* __restrict__ deg) {
  int i = blockIdx.x * 256 + threadIdx.x;
  if (i < NN) {
    float d = deg[i];
    deg[i] = (d > 0.0f) ? rsqrtf(d) : 0.0f;
  }
}

// ---------------- w_off = -w * dinv[src] * dinv[dst] ----------------
__global__ __launch_bounds__(256) void woff_k(const long long* __restrict__ src,
                                              const long long* __restrict__ dst,
                                              const float* __restrict__ w,
                                              const float* __restrict__ dinv,
                                              float* __restrict__ wof) {
  int e = blockIdx.x * 256 + threadIdx.x;
  if (e < EE) wof[e] = -w[e] * dinv[(int)src[e]] * dinv[(int)dst[e]];
}

// ---------------- Yout[dst] += scale * w_off * Xin[src]  (4 channels/thread) ----------------
__global__ __launch_bounds__(256) void spmm_k(const long long* __restrict__ src,
                                              const long long* __restrict__ dst,
                                              const float* __restrict__ wof,
                                              const float* __restrict__ Xin,
                                              float* __restrict__ Yout,
                                              float scale) {
  int t = blockIdx.x * 256 + threadIdx.x;   // EE*16 threads
  int e = t >> 4;
  if (e >= EE) return;
  int c = (t & 15) << 2;
  float wv = scale * wof[e];
  int s = (int)src[e];
  int d = (int)dst[e];
  const float4 x = *(const float4*)(Xin + (size_t)s * CIN + c);
  float* y = Yout + (size_t)d * CIN + c;
  atomicAdd(y + 0, wv * x.x);
  atomicAdd(y + 1, wv * x.y);
  atomicAdd(y + 2, wv * x.z);
  atomicAdd(y + 3, wv * x.w);
}

// ---------------- fused WMMA GEMM + GRU gate epilogue ----------------
// grid.x = NN/16 row tiles, 128 threads = 4 wave32, wave w owns col tile [16w,16w+16)
// out = (1 - sigmoid(sum_k Txk@Wz[k] + bxz + bhz)) * tanh(sum_k Txk@Wh[k] + bxh + bhh)
__global__ __launch_bounds__(128) void gemm_gate_k(
    const float* __restrict__ Tx0, const float* __restrict__ Tx1,
    const float* __restrict__ Tx2,
    const float* __restrict__ Wz, const float* __restrict__ Wh,
    const float* __restrict__ bxz, const float* __restrict__ bhz,
    const float* __restrict__ bxh, const float* __restrict__ bhh,
    float* __restrict__ out) {
  const int lane = threadIdx.x & 31;
  const int wave = threadIdx.x >> 5;
  const int row0 = blockIdx.x << 4;
  const int col0 = wave << 4;
  const int m  = lane & 15;   // A row / B col / D col within tile
  const int hi = lane >> 4;   // upper half-wave -> K offset +2 (A/B), row offset +8 (D)

  v8f accz = {};
  v8f acch = {};

  const float* Txk[3] = {Tx0, Tx1, Tx2};
  for (int k = 0; k < 3; ++k) {
    // A-frag base: Txk[row0+m][hi*2 + 4*kk]  (16x4 f32 A layout)
    const float* A  = Txk[k] + (size_t)(row0 + m) * CIN + hi * 2;
    // B-frag base: W[k][hi*2 + 4*kk][col0+m] (4x16 f32 B layout, row stride 64)
    const float* Bz = Wz + (size_t)k * CIN * 64 + (hi * 2) * 64 + col0 + m;
    const float* Bh = Wh + (size_t)k * CIN * 64 + (hi * 2) * 64 + col0 + m;
#pragma unroll
    for (int kk = 0; kk < 16; ++kk) {
      v2f a = *(const v2f*)(A + kk * 4);          // 8B aligned: (64*row + even) floats
      v2f bz, bh;
      bz.x = Bz[kk * 4 * 64];
      bz.y = Bz[kk * 4 * 64 + 64];
      bh.x = Bh[kk * 4 * 64];
      bh.y = Bh[kk * 4 * 64 + 64];
      accz = __builtin_amdgcn_wmma_f32_16x16x4_f32(false, a, false, bz,
                                                   (short)0, accz, false, false);
      acch = __builtin_amdgcn_wmma_f32_16x16x4_f32(false, a, false, bh,
                                                   (short)0, acch, false, false);
    }
  }

  // D layout: VGPR v -> row = row0 + v + 8*hi, col = col0 + (lane&15)
#pragma unroll
  for (int v = 0; v < 8; ++v) {
    int row = row0 + v + 8 * hi;
    int col = col0 + m;
    float zf = accz[v] + bxz[col] + bhz[col];
    float hf = acch[v] + bxh[col] + bhh[col];
    float sig = 1.0f / (1.0f + expf(-zf));
    out[(size_t)row * 64 + col] = (1.0f - sig) * tanhf(hf);
  }
}

extern "C" void kernel_launch(void* const* d_in, const int* in_sizes, int n_in,
                              void* d_out, int out_size, void* d_ws, size_t ws_size,
                              hipStream_t stream) {
  (void)in_sizes; (void)n_in; (void)out_size; (void)ws_size;

  const float*     X   = (const float*)d_in[0];
  const long long* ei  = (const long long*)d_in[1];   // (2, E) int64
  const float*     ew  = (const float*)d_in[2];
  const float*     Wxz = (const float*)d_in[3];
  const float*     Wxh = (const float*)d_in[7];
  const float*     bxz = (const float*)d_in[9];
  const float*     bhz = (const float*)d_in[10];
  const float*     bxh = (const float*)d_in[13];
  const float*     bhh = (const float*)d_in[14];
  float*           out = (float*)d_out;

  const long long* src = ei;
  const long long* dst = ei + EE;

  float* Tx1 = (float*)d_ws;                       // N*64
  float* Tx2 = Tx1 + (size_t)NN * CIN;             // N*64
  float* deg = Tx2 + (size_t)NN * CIN;             // N   (reused as d^{-1/2})
  float* wof = deg + NN;                           // E

  init_k  <<<(NN * CIN + 255) / 256, 256, 0, stream>>>(X, Tx1, Tx2, deg);
  degree_k<<<(EE + 255) / 256,       256, 0, stream>>>(src, ew, deg);
  dinv_k  <<<(NN + 255) / 256,       256, 0, stream>>>(deg);
  woff_k  <<<(EE + 255) / 256,       256, 0, stream>>>(src, dst, ew, deg, wof);

  // Tx1 = L_hat @ X
  spmm_k<<<(EE * 16 + 255) / 256, 256, 0, stream>>>(src, dst, wof, X,   Tx1, 1.0f);
  // Tx2 = 2 * L_hat @ Tx1 - X   (Tx2 pre-initialized to -X)
  spmm_k<<<(EE * 16 + 255) / 256, 256, 0, stream>>>(src, dst, wof, Tx1, Tx2, 2.0f);

  gemm_gate_k<<<NN / 16, 128, 0, stream>>>(X, Tx1, Tx2, Wxz, Wxh,
                                           bxz, bhz, bxh, bhh, out);
}